// MoEGate_35476429865152
// MI455X (gfx1250) — compile-verified
//
#include <hip/hip_runtime.h>
#include <hip/hip_bf16.h>

typedef __attribute__((ext_vector_type(2))) float v2f;
typedef __attribute__((ext_vector_type(8))) float v8f;

#define HIDDEN   4096
#define N_EXP    64
#define N_GROUP  8
#define EPG      8     // experts per group
#define TOPK_G   3
#define TOP_K    6

// Device-pass-only check: __has_builtin for amdgcn builtins is false on the
// host pass, which is fine -- the host pass never codegens the kernel body.
#if defined(__HIP_DEVICE_COMPILE__) && !__has_builtin(__builtin_amdgcn_wmma_f32_16x16x4_f32)
#error "wmma_f32_16x16x4_f32 builtin not available on this device toolchain"
#endif

// ---------------------------------------------------------------------------
// Kernel A: logits[T][64] = x[T][4096] * weight[64][4096]^T  via
// V_WMMA_F32_16X16X4_F32.  One wave per 16-token tile; 4 accumulator tiles
// cover all 64 experts so every B (weight) fragment load feeds 1 WMMA while
// the A fragment feeds 4.
// ---------------------------------------------------------------------------
__global__ __launch_bounds__(256)
void moe_logits_wmma_kernel(const float* __restrict__ x,
                            const float* __restrict__ w,
                            float* __restrict__ logits,
                            int T)
{
    const int lane = threadIdx.x & 31;
    const int wave = threadIdx.x >> 5;
    const int tile = blockIdx.x * (blockDim.x >> 5) + wave;   // 16-token tile id
    const int row0 = tile * 16;
    if (row0 >= T) return;                 // wave-uniform: T % 16 == 0, EXEC stays all-ones

    // f32 A/B fragment layout (ISA 7.12.2): lanes 0-15 hold K=0,1; lanes 16-31 hold K=2,3.
    const int m  = lane & 15;              // row within tile (token / expert)
    const int kk = (lane >> 4) * 2;        // K sub-offset: 0 or 2

    const float* __restrict__ xrow  = x + (size_t)(row0 + m) * HIDDEN + kk;
    const float* __restrict__ wrow0 = w + (size_t)( 0 + m) * HIDDEN + kk;
    const float* __restrict__ wrow1 = w + (size_t)(16 + m) * HIDDEN + kk;
    const float* __restrict__ wrow2 = w + (size_t)(32 + m) * HIDDEN + kk;
    const float* __restrict__ wrow3 = w + (size_t)(48 + m) * HIDDEN + kk;

    v8f c0 = {}, c1 = {}, c2 = {}, c3 = {};

    for (int kb = 0; kb < HIDDEN; kb += 64) {
        // prefetch the streaming activation row ~1KB ahead (global_prefetch_b8)
        __builtin_prefetch(xrow + kb + 256, 0, 1);
        #pragma unroll
        for (int ku = 0; ku < 64; ku += 4) {
            const int k = kb + ku;
            v2f a  = *(const v2f*)(xrow  + k);
            v2f b0 = *(const v2f*)(wrow0 + k);
            v2f b1 = *(const v2f*)(wrow1 + k);
            v2f b2 = *(const v2f*)(wrow2 + k);
            v2f b3 = *(const v2f*)(wrow3 + k);
            c0 = __builtin_amdgcn_wmma_f32_16x16x4_f32(false, a, false, b0, (short)0, c0, false, false);
            c1 = __builtin_amdgcn_wmma_f32_16x16x4_f32(false, a, false, b1, (short)0, c1, false, false);
            c2 = __builtin_amdgcn_wmma_f32_16x16x4_f32(false, a, false, b2, (short)0, c2, false, false);
            c3 = __builtin_amdgcn_wmma_f32_16x16x4_f32(false, a, false, b3, (short)0, c3, false, false);
        }
    }

    // C/D layout: VGPR r -> lanes 0-15: (M=r, N=lane), lanes 16-31: (M=r+8, N=lane-16)
    const int nlo = lane & 15;
    const int mhi = (lane >> 4) * 8;
    float* __restrict__ out = logits + (size_t)row0 * N_EXP;
    #pragma unroll
    for (int r = 0; r < 8; ++r) {
        const size_t base = (size_t)(r + mhi) * N_EXP;
        out[base +  0 + nlo] = c0[r];
        out[base + 16 + nlo] = c1[r];
        out[base + 32 + nlo] = c2[r];
        out[base + 48 + nlo] = c3[r];
    }
}

// ---------------------------------------------------------------------------
// Kernel B: softmax over 64 experts, group-limited (top-3 of 8 groups by
// group max), then top-6 experts.  One thread per token; all 64 scores live
// in registers (fully unrolled, bitmask bookkeeping, no dynamic indexing).
// ---------------------------------------------------------------------------
__global__ __launch_bounds__(256)
void moe_gate_kernel(const float* __restrict__ logits,
                     int*   __restrict__ out_idx,
                     float* __restrict__ out_w,
                     int T)
{
    const int t = blockIdx.x * blockDim.x + threadIdx.x;
    if (t >= T) return;

    const float* __restrict__ s = logits + (size_t)t * N_EXP;

    float v[N_EXP];
    float mx = -3.402823466e+38f;
    #pragma unroll
    for (int i = 0; i < N_EXP; ++i) { v[i] = s[i]; mx = fmaxf(mx, v[i]); }

    float sum = 0.0f;
    #pragma unroll
    for (int i = 0; i < N_EXP; ++i) { v[i] = __expf(v[i] - mx); sum += v[i]; }
    const float inv = __frcp_rn(sum);
    #pragma unroll
    for (int i = 0; i < N_EXP; ++i) v[i] *= inv;

    // per-group max
    float gm[N_GROUP];
    #pragma unroll
    for (int g = 0; g < N_GROUP; ++g) {
        float bm = v[g * EPG];
        #pragma unroll
        for (int j = 1; j < EPG; ++j) bm = fmaxf(bm, v[g * EPG + j]);
        gm[g] = bm;
    }

    // top-3 groups (strict > keeps lowest index on ties, like jax top_k)
    unsigned gsel = 0u;
    #pragma unroll
    for (int pick = 0; pick < TOPK_G; ++pick) {
        float best = -1.0f; int bg = 0;
        #pragma unroll
        for (int g = 0; g < N_GROUP; ++g) {
            const bool taken = (gsel >> g) & 1u;
            if (!taken && gm[g] > best) { best = gm[g]; bg = g; }
        }
        gsel |= 1u << bg;
    }

    unsigned long long emask = 0ull;
    #pragma unroll
    for (int g = 0; g < N_GROUP; ++g)
        if ((gsel >> g) & 1u) emask |= (0xFFull << (g * EPG));

    // top-6 experts among selected groups, descending
    unsigned long long chosen = 0ull;
    #pragma unroll
    for (int pick = 0; pick < TOP_K; ++pick) {
        float best = -1.0f; int be = 0;
        #pragma unroll
        for (int i = 0; i < N_EXP; ++i) {
            const bool live = ((emask >> i) & 1ull) && !((chosen >> i) & 1ull);
            const float val = live ? v[i] : -1.0f;
            if (val > best) { best = val; be = i; }
        }
        chosen |= 1ull << be;
        out_idx[(size_t)t * TOP_K + pick] = be;
        out_w  [(size_t)t * TOP_K + pick] = best;
    }
}

// ---------------------------------------------------------------------------
extern "C" void kernel_launch(void* const* d_in, const int* in_sizes, int n_in,
                              void* d_out, int out_size, void* d_ws, size_t ws_size,
                              hipStream_t stream)
{
    const float* x = (const float*)d_in[0];   // (B,S,H) fp32, flat T*H
    const float* w = (const float*)d_in[1];   // (E,H)   fp32

    const int T = in_sizes[0] / HIDDEN;       // 16384 tokens

    float* logits  = (float*)d_ws;                      // T * 64 fp32 scratch (4 MB)
    int*   out_idx = (int*)d_out;                       // first output: int32 T*6
    float* out_w   = (float*)d_out + (size_t)T * TOP_K; // second output: fp32 T*6

    // Kernel A: 8 waves/block * 16 tokens/wave = 128 tokens per block
    const int tokens_per_block = 128;
    dim3 blockA(256), gridA((T + tokens_per_block - 1) / tokens_per_block);
    moe_logits_wmma_kernel<<<gridA, blockA, 0, stream>>>(x, w, logits, T);

    // Kernel B: one thread per token
    dim3 blockB(256), gridB((T + 255) / 256);
    moe_gate_kernel<<<gridB, blockB, 0, stream>>>(logits, out_idx, out_w, T);
}